// BiomassGNN_61418032333218
// MI455X (gfx1250) — compile-verified
//
#include <hip/hip_runtime.h>

// ---------------- problem constants (match reference) ----------------
#define N_NODES   100000
#define N_EDGES   1200000
#define D_INF     7
#define HDIM      64
#define G_GRAPHS  64
#define L_LAYERS  2
#define BN_EPS    1e-5f

typedef __attribute__((ext_vector_type(2))) float v2f;
typedef __attribute__((ext_vector_type(8))) float v8f;

// ================= CSR build =================
__global__ void zero_cnt_kernel(int* __restrict__ cnt) {
    int i = blockIdx.x * blockDim.x + threadIdx.x;
    if (i < N_NODES) cnt[i] = 0;
}

__global__ void count_kernel(const int* __restrict__ dst, int* __restrict__ cnt) {
    int i = blockIdx.x * blockDim.x + threadIdx.x;
    if (i < N_EDGES) atomicAdd(&cnt[dst[i]], 1);
}

__global__ void dinv_kernel(const int* __restrict__ cnt, float* __restrict__ dinv) {
    int i = blockIdx.x * blockDim.x + threadIdx.x;
    if (i < N_NODES) dinv[i] = rsqrtf(1.0f + (float)cnt[i]);  // +1 for self-loop
}

// single-block two-phase exclusive scan: row_ptr[i] = sum(cnt[0..i)), cursor copy
__global__ __launch_bounds__(1024)
void scan_kernel(const int* __restrict__ cnt,
                 int* __restrict__ row_ptr,
                 int* __restrict__ cursor) {
    __shared__ int lsum[1024];
    const int t = threadIdx.x;
    const int chunk = (N_NODES + 1023) / 1024;   // 98
    const int begin = t * chunk;
    const int end   = min(begin + chunk, N_NODES);
    int s = 0;
    for (int i = begin; i < end; ++i) s += cnt[i];
    lsum[t] = s;
    __syncthreads();
    // Hillis-Steele inclusive scan over 1024 partials
    for (int off = 1; off < 1024; off <<= 1) {
        int v = (t >= off) ? lsum[t - off] : 0;
        __syncthreads();
        lsum[t] += v;
        __syncthreads();
    }
    int run = (t == 0) ? 0 : lsum[t - 1];
    for (int i = begin; i < end; ++i) {
        row_ptr[i] = run;
        cursor[i]  = run;
        run += cnt[i];
    }
}

// csr[pos] = {src, bits(dinv[src]*dinv[dst])}
__global__ void fill_csr_kernel(const int* __restrict__ src,
                                const int* __restrict__ dst,
                                const float* __restrict__ dinv,
                                int* __restrict__ cursor,
                                int2* __restrict__ csr) {
    int e = blockIdx.x * blockDim.x + threadIdx.x;
    if (e >= N_EDGES) return;
    int s = src[e];
    int d = dst[e];
    int pos = atomicAdd(&cursor[d], 1);
    int2 v;
    v.x = s;
    v.y = __float_as_int(dinv[s] * dinv[d]);
    csr[pos] = v;
}

// ================= input projection: h = relu(x @ W_in + b_in) =================
__global__ void input_proj_kernel(const float* __restrict__ x,
                                  const float* __restrict__ Win,
                                  const float* __restrict__ bin,
                                  float* __restrict__ h) {
    int tid = blockIdx.x * blockDim.x + threadIdx.x;
    if (tid >= N_NODES * HDIM) return;
    int n = tid >> 6;
    int j = tid & 63;
    float acc = bin[j];
#pragma unroll
    for (int k = 0; k < D_INF; ++k)
        acc += x[n * D_INF + k] * Win[k * HDIM + j];
    h[tid] = fmaxf(acc, 0.0f);
}

// ================= WMMA GEMM: C[N,64] = A[N,64] @ W[64,64] (fp32) =================
// One wave per 16x64 output tile; V_WMMA_F32_16X16X4_F32 x 64 (16 K-steps, 4 N-tiles).
__global__ __launch_bounds__(128)
void gemm_wmma_kernel(const float* __restrict__ A,
                      const float* __restrict__ W,
                      float* __restrict__ C) {
    const int lane  = threadIdx.x & 31;
    const int wave  = threadIdx.x >> 5;
    const int row_base = (blockIdx.x * 4 + wave) * 16;
    if (row_base >= N_NODES) return;          // wave-uniform: EXEC stays all-1s

    const int m     = lane & 15;
    const int khalf = lane >> 4;              // 0 or 1

    v8f acc[4];
#pragma unroll
    for (int nt = 0; nt < 4; ++nt) acc[nt] = (v8f){};

    const float* arow = A + (size_t)(row_base + m) * HDIM;

#pragma unroll
    for (int ko = 0; ko < HDIM; ko += 4) {
        const int krow = ko + khalf * 2;
        v2f a = *(const v2f*)(arow + krow);   // A[m][krow], A[m][krow+1]
#pragma unroll
        for (int nt = 0; nt < 4; ++nt) {
            v2f b;
            b.x = W[(krow + 0) * HDIM + nt * 16 + m];
            b.y = W[(krow + 1) * HDIM + nt * 16 + m];
            acc[nt] = __builtin_amdgcn_wmma_f32_16x16x4_f32(
                false, a, false, b, (short)0, acc[nt], false, false);
        }
    }

#pragma unroll
    for (int nt = 0; nt < 4; ++nt) {
#pragma unroll
        for (int r = 0; r < 8; ++r) {
            int row = row_base + r + khalf * 8;
            C[(size_t)row * HDIM + nt * 16 + m] = acc[nt][r];
        }
    }
}

// ================= fused pull-aggregate + bias + BN + relu + residual =================
// One wave per node; lane owns columns {2*lane, 2*lane+1} -> all hw gathers are
// coalesced 256B wave-wide b64 accesses. Zero atomics.
__global__ __launch_bounds__(256)
void aggregate_bn_kernel(const float* __restrict__ hw,
                         const float* __restrict__ dinv,
                         const int* __restrict__ row_ptr,
                         const int* __restrict__ cnt,
                         const int2* __restrict__ csr,
                         const float* __restrict__ cb,
                         const float* __restrict__ gamma,
                         const float* __restrict__ beta,
                         const float* __restrict__ mean,
                         const float* __restrict__ var,
                         float* __restrict__ h,
                         int residual) {
    const int wave = threadIdx.x >> 5;
    const int lane = threadIdx.x & 31;
    const int n = blockIdx.x * 8 + wave;
    if (n >= N_NODES) return;                  // wave-uniform

    const int j = 2 * lane;
    float d = dinv[n];
    v2f selfv = *(const v2f*)(hw + (size_t)n * HDIM + j);
    float a0 = selfv.x * d * d;
    float a1 = selfv.y * d * d;

    const int beg = row_ptr[n];
    const int num = cnt[n];
    for (int k = 0; k < num; ++k) {
        int2 ed = csr[beg + k];                // uniform across wave -> 1 request
        float w = __int_as_float(ed.y);
        v2f hv = *(const v2f*)(hw + (size_t)ed.x * HDIM + j);
        a0 += hv.x * w;
        a1 += hv.y * w;
    }

    // bias + BN(eval) + relu
    a0 = (a0 + cb[j]     - mean[j])     * rsqrtf(var[j]     + BN_EPS) * gamma[j]     + beta[j];
    a1 = (a1 + cb[j + 1] - mean[j + 1]) * rsqrtf(var[j + 1] + BN_EPS) * gamma[j + 1] + beta[j + 1];
    a0 = fmaxf(a0, 0.0f);
    a1 = fmaxf(a1, 0.0f);

    float* hp = h + (size_t)n * HDIM + j;
    v2f r;
    if (residual) {
        v2f old = *(const v2f*)hp;
        r.x = old.x + a0;
        r.y = old.y + a1;
    } else {
        r.x = a0;
        r.y = a1;
    }
    *(v2f*)hp = r;
}

// ================= pooling =================
__global__ void pool_zero_kernel(float* __restrict__ sums, float* __restrict__ cnt) {
    int i = blockIdx.x * blockDim.x + threadIdx.x;
    if (i < G_GRAPHS * HDIM) sums[i] = 0.0f;
    else if (i < G_GRAPHS * HDIM + G_GRAPHS) cnt[i - G_GRAPHS * HDIM] = 0.0f;
}

#define POOL_NODES_PER_BLOCK 1024
__global__ __launch_bounds__(256)
void pool_lds_kernel(const float* __restrict__ h,
                     const int* __restrict__ batch,
                     float* __restrict__ sums,
                     float* __restrict__ cnt) {
    __shared__ float ls[G_GRAPHS * HDIM];      // 16 KB
    __shared__ float lc[G_GRAPHS];
    const int t = threadIdx.x;
    for (int i = t; i < G_GRAPHS * HDIM; i += 256) ls[i] = 0.0f;
    if (t < G_GRAPHS) lc[t] = 0.0f;
    __syncthreads();

    const int base = blockIdx.x * POOL_NODES_PER_BLOCK;
    const int lim  = min(base + POOL_NODES_PER_BLOCK, N_NODES);
    const int j = t & 63;
    for (int n = base + (t >> 6); n < lim; n += 4) {
        int g = batch[n];
        atomicAdd(&ls[g * HDIM + j], h[(size_t)n * HDIM + j]);  // ds_add_f32, bank-conflict-free
        if (j == 0) atomicAdd(&lc[g], 1.0f);
    }
    __syncthreads();
    for (int i = t; i < G_GRAPHS * HDIM; i += 256) atomicAdd(&sums[i], ls[i]);
    if (t < G_GRAPHS) atomicAdd(&cnt[t], lc[t]);
}

// ================= MLP head =================
__global__ void mlp_kernel(const float* __restrict__ sums,
                           const float* __restrict__ cnt,
                           const float* __restrict__ W1,
                           const float* __restrict__ b1,
                           const float* __restrict__ W2,
                           const float* __restrict__ b2,
                           float* __restrict__ out) {
    int g = threadIdx.x;
    if (g >= G_GRAPHS) return;
    float inv = 1.0f / fmaxf(cnt[g], 1.0f);
    float p[HDIM];
#pragma unroll
    for (int j = 0; j < HDIM; ++j) p[j] = sums[g * HDIM + j] * inv;
    float o = b2[0];
    for (int k = 0; k < HDIM / 2; ++k) {
        float hk = b1[k];
#pragma unroll
        for (int j = 0; j < HDIM; ++j) hk += p[j] * W1[j * (HDIM / 2) + k];
        o += fmaxf(hk, 0.0f) * W2[k];
    }
    out[g] = o;
}

// ================= launcher =================
extern "C" void kernel_launch(void* const* d_in, const int* in_sizes, int n_in,
                              void* d_out, int out_size, void* d_ws, size_t ws_size,
                              hipStream_t stream) {
    const float* x     = (const float*)d_in[0];
    const int*   edge  = (const int*)  d_in[1];
    const int*   batch = (const int*)  d_in[2];
    const float* Win   = (const float*)d_in[3];
    const float* bin   = (const float*)d_in[4];
    const float* convW = (const float*)d_in[5];   // [L,H,H]
    const float* convb = (const float*)d_in[6];   // [L,H]
    const float* gam   = (const float*)d_in[7];
    const float* bet   = (const float*)d_in[8];
    const float* mea   = (const float*)d_in[9];
    const float* var   = (const float*)d_in[10];
    const float* W1    = (const float*)d_in[11];
    const float* b1    = (const float*)d_in[12];
    const float* W2    = (const float*)d_in[13];
    const float* b2    = (const float*)d_in[14];
    float*       out   = (float*)d_out;

    const int* src = edge;             // edge_index[0]
    const int* dst = edge + N_EDGES;   // edge_index[1]

    const size_t NH = (size_t)N_NODES * HDIM;
    float* h       = (float*)d_ws;
    float* hw      = h + NH;
    int2*  csr     = (int2*)(hw + NH);                 // 8B-aligned (offset = 51.2MB)
    float* dinv    = (float*)(csr + N_EDGES);
    int*   cnt_i   = (int*)(dinv + N_NODES);
    int*   row_ptr = cnt_i + N_NODES;
    int*   cursor  = row_ptr + N_NODES;
    float* sums    = (float*)(cursor + N_NODES);
    float* cntf    = sums + (size_t)G_GRAPHS * HDIM;

    const int T = 256;
    const int nhBlocks   = (int)((NH + T - 1) / T);
    const int nBlocks    = (N_NODES + T - 1) / T;
    const int eBlocks    = (N_EDGES + T - 1) / T;
    const int gemmBlocks = (N_NODES + 63) / 64;
    const int aggBlocks  = (N_NODES + 7) / 8;
    const int poolBlocks = (N_NODES + POOL_NODES_PER_BLOCK - 1) / POOL_NODES_PER_BLOCK;

    // ---- CSR build (once; edge structure is layer-independent) ----
    zero_cnt_kernel<<<nBlocks, T, 0, stream>>>(cnt_i);
    count_kernel<<<eBlocks, T, 0, stream>>>(dst, cnt_i);
    dinv_kernel<<<nBlocks, T, 0, stream>>>(cnt_i, dinv);
    scan_kernel<<<1, 1024, 0, stream>>>(cnt_i, row_ptr, cursor);
    fill_csr_kernel<<<eBlocks, T, 0, stream>>>(src, dst, dinv, cursor, csr);

    // ---- input projection ----
    input_proj_kernel<<<nhBlocks, T, 0, stream>>>(x, Win, bin, h);

    // ---- GCN layers: WMMA GEMM + fused pull-aggregate/BN/relu/residual ----
    for (int l = 0; l < L_LAYERS; ++l) {
        gemm_wmma_kernel<<<gemmBlocks, 128, 0, stream>>>(h, convW + (size_t)l * HDIM * HDIM, hw);
        aggregate_bn_kernel<<<aggBlocks, T, 0, stream>>>(
            hw, dinv, row_ptr, cnt_i, csr,
            convb + l * HDIM, gam + l * HDIM, bet + l * HDIM,
            mea + l * HDIM, var + l * HDIM, h, l > 0 ? 1 : 0);
    }

    // ---- global mean pool (LDS-staged) + MLP head ----
    pool_zero_kernel<<<(G_GRAPHS * HDIM + G_GRAPHS + T - 1) / T, T, 0, stream>>>(sums, cntf);
    pool_lds_kernel<<<poolBlocks, T, 0, stream>>>(h, batch, sums, cntf);
    mlp_kernel<<<1, G_GRAPHS, 0, stream>>>(sums, cntf, W1, b1, W2, b2, out);
}